// SSM_42872363549112
// MI455X (gfx1250) — compile-verified
//
#include <hip/hip_runtime.h>
#include <cstdint>

// Problem dims (fixed by reference)
#define BSZ   2
#define DM    1536
#define LSEQ  2048
#define NST   16
#define NC    32            // chunks
#define TC    64            // timesteps per chunk
#define DBLKS (DM / 16)     // 96 blocks of 16 d-channels
#define WPB   4             // waves per workgroup
#define ROWP  68            // padded LDS row stride (floats): 272B = 17*16B, bank-conflict free
#define TILE  (16 * ROWP)   // 1088 floats per 16x64 tile

typedef float v2f __attribute__((ext_vector_type(2)));
typedef float v8f __attribute__((ext_vector_type(8)));

// ---- CDNA5 async global->LDS copy (ASYNCcnt-tracked) ------------------------
__device__ __forceinline__ void async_b128(uint32_t lds_off, const float* g) {
    // VDST = VGPR with LDS byte address, VADDR = 64-bit global address
    asm volatile("global_load_async_to_lds_b128 %0, %1, off"
                 :: "v"(lds_off), "v"(g) : "memory");
}
__device__ __forceinline__ void wait_async0() {
    asm volatile("s_wait_asynccnt 0" ::: "memory");
}
__device__ __forceinline__ uint32_t lds_off_of(const float* p) {
    // generic LDS pointer: low 32 bits are the wave-relative LDS offset
    return (uint32_t)(uint64_t)p;
}

// =============================================================================
// Phase A: per-chunk aggregates.  For each (b, d, n, chunk):
//   aProd = prod_t exp(delta*A),  hLocal = chunk scan end-state from h=0
// =============================================================================
__global__ __launch_bounds__(128)
void ssm_phaseA(const float* __restrict__ x, const float* __restrict__ delta,
                const float* __restrict__ A, const float* __restrict__ Bm,
                float* __restrict__ aBuf, float* __restrict__ hBuf) {
    __shared__ __align__(16) float smem[TILE * (1 + 2 * WPB)]; // bT | xT[4] | dT[4]
    const int tid  = threadIdx.x;
    const int lane = tid & 31;
    const int wv   = tid >> 5;
    const int c    = blockIdx.x;
    const int b    = blockIdx.z;
    const int dblk = blockIdx.y * WPB + wv;
    const int d0   = dblk * 16;
    const int cbase = c * TC;
    const int row  = lane & 15;
    const int hi   = lane >> 4;

    // B tile (shared): 256 b128 segments over 128 threads
    #pragma unroll
    for (int it = 0; it < 2; ++it) {
        int si = it * 128 + tid;
        int r = si >> 4, s = si & 15;
        async_b128(lds_off_of(&smem[r * ROWP + s * 4]),
                   Bm + (size_t)(b * NST + r) * LSEQ + cbase + s * 4);
    }
    // per-wave x / delta tiles: 256 segments over 32 lanes each
    float* xT = smem + TILE * (1 + wv);
    float* dT = smem + TILE * (1 + WPB + wv);
    #pragma unroll
    for (int it = 0; it < 8; ++it) {
        int si = it * 32 + lane;
        int r = si >> 4, s = si & 15;
        size_t gi = (size_t)(b * DM + d0 + r) * LSEQ + cbase + s * 4;
        async_b128(lds_off_of(&xT[r * ROWP + s * 4]), x + gi);
        async_b128(lds_off_of(&dT[r * ROWP + s * 4]), delta + gi);
    }
    wait_async0();
    __syncthreads();

    // per-lane state layout == WMMA A-matrix layout: n = 4*(q>>1) + 2*hi + (q&1)
    const int d = d0 + row;
    float Al2[8]; int noff[8];
    #pragma unroll
    for (int q = 0; q < 8; ++q) {
        int n = 4 * (q >> 1) + 2 * hi + (q & 1);
        Al2[q]  = A[d * NST + n] * 1.44269504088896340736f; // pre-scale for exp2
        noff[q] = n * ROWP;
    }
    float h[8] = {0, 0, 0, 0, 0, 0, 0, 0};
    float p[8] = {1, 1, 1, 1, 1, 1, 1, 1};
    #pragma unroll 4
    for (int t = 0; t < TC; ++t) {
        float dv = dT[row * ROWP + t];
        float xv = xT[row * ROWP + t];
        float dx = dv * xv;
        // Batched: 8 independent muls -> 8 independent exps -> 8 FMAs.
        // Keeps >=7 independent instrs between each v_exp_f32 and its consumer,
        // avoiding TRANS32 stalls / v_nops.
        float e[8], u[8];
        #pragma unroll
        for (int q = 0; q < 8; ++q) e[q] = dv * Al2[q];
        #pragma unroll
        for (int q = 0; q < 8; ++q) e[q] = __builtin_amdgcn_exp2f(e[q]);
        #pragma unroll
        for (int q = 0; q < 8; ++q) u[q] = dx * smem[noff[q] + t];   // bT at offset 0
        #pragma unroll
        for (int q = 0; q < 8; ++q) {
            h[q] = __builtin_fmaf(e[q], h[q], u[q]);
            p[q] *= e[q];
        }
    }
    size_t slot = (((size_t)((b * DBLKS + dblk) * NC + c)) << 8) + lane * 8;
    #pragma unroll
    for (int q = 0; q < 8; ++q) { aBuf[slot + q] = p[q]; hBuf[slot + q] = h[q]; }
}

// =============================================================================
// Phase B: inter-chunk scan over the 32 chunk aggregates per (b,d,n).
// Writes incoming state hIn[c] in-place over aBuf.
// =============================================================================
__global__ __launch_bounds__(256)
void ssm_phaseB(float* __restrict__ aBuf, const float* __restrict__ hBuf) {
    int tid = blockIdx.x * 256 + threadIdx.x;     // 49152 threads
    int e   = tid & 255;
    int bd  = tid >> 8;                            // (b, dblk) pair, 0..191
    size_t base = ((size_t)bd * NC << 8) + e;
    float h = 0.f;
    for (int c = 0; c < NC; ++c) {
        size_t i = base + ((size_t)c << 8);
        float aP = aBuf[i];
        float hL = hBuf[i];
        aBuf[i] = h;                               // hIn for chunk c
        h = __builtin_fmaf(aP, h, hL);
    }
}

// =============================================================================
// Phase C: replay chunk with correct incoming state; WMMA readout; final output.
// =============================================================================
__global__ __launch_bounds__(128)
void ssm_phaseC(const float* __restrict__ x, const float* __restrict__ delta,
                const float* __restrict__ A, const float* __restrict__ Bm,
                const float* __restrict__ Cm, const float* __restrict__ Dp,
                const float* __restrict__ z, const float* __restrict__ hIn,
                float* __restrict__ out) {
    __shared__ __align__(16) float smem[TILE * (2 + 2 * WPB)]; // bT | cT | xT[4] | dT[4]
    const int tid  = threadIdx.x;
    const int lane = tid & 31;
    const int wv   = tid >> 5;
    const int c    = blockIdx.x;
    const int b    = blockIdx.z;
    const int dblk = blockIdx.y * WPB + wv;
    const int d0   = dblk * 16;
    const int cbase = c * TC;
    const int row  = lane & 15;
    const int hi   = lane >> 4;

    float* cT = smem + TILE;
    #pragma unroll
    for (int it = 0; it < 2; ++it) {
        int si = it * 128 + tid;
        int r = si >> 4, s = si & 15;
        size_t gi = (size_t)(b * NST + r) * LSEQ + cbase + s * 4;
        async_b128(lds_off_of(&smem[r * ROWP + s * 4]), Bm + gi);
        async_b128(lds_off_of(&cT[r * ROWP + s * 4]),   Cm + gi);
    }
    float* xT = smem + TILE * (2 + wv);
    float* dT = smem + TILE * (2 + WPB + wv);
    #pragma unroll
    for (int it = 0; it < 8; ++it) {
        int si = it * 32 + lane;
        int r = si >> 4, s = si & 15;
        size_t gi = (size_t)(b * DM + d0 + r) * LSEQ + cbase + s * 4;
        async_b128(lds_off_of(&xT[r * ROWP + s * 4]), x + gi);
        async_b128(lds_off_of(&dT[r * ROWP + s * 4]), delta + gi);
    }
    wait_async0();
    __syncthreads();

    const int d = d0 + row;
    float Al2[8]; int noff[8];
    #pragma unroll
    for (int q = 0; q < 8; ++q) {
        int n = 4 * (q >> 1) + 2 * hi + (q & 1);
        Al2[q]  = A[d * NST + n] * 1.44269504088896340736f;
        noff[q] = n * ROWP;
    }
    float Dv[8];
    #pragma unroll
    for (int r = 0; r < 8; ++r) Dv[r] = Dp[d0 + r + 8 * hi];

    size_t slot = (((size_t)((b * DBLKS + dblk) * NC + c)) << 8) + lane * 8;
    float h[8];
    #pragma unroll
    for (int q = 0; q < 8; ++q) h[q] = hIn[slot + q];

    const size_t obase = (size_t)b * DM * LSEQ;

    for (int sub = 0; sub < 4; ++sub) {
        v8f acc0 = {0, 0, 0, 0, 0, 0, 0, 0};
        v8f acc1 = {0, 0, 0, 0, 0, 0, 0, 0};
        #pragma unroll
        for (int j = 0; j < 16; ++j) {
            const int t = sub * 16 + j;
            float dv = dT[row * ROWP + t];
            float xv = xT[row * ROWP + t];
            float dx = dv * xv;
            // Batched recurrence update (see phase A comment).
            float e[8], u[8], cv[8];
            #pragma unroll
            for (int q = 0; q < 8; ++q) e[q] = dv * Al2[q];
            #pragma unroll
            for (int q = 0; q < 8; ++q) e[q] = __builtin_amdgcn_exp2f(e[q]);
            #pragma unroll
            for (int q = 0; q < 8; ++q) u[q] = dx * smem[noff[q] + t];  // bT
            #pragma unroll
            for (int q = 0; q < 8; ++q) cv[q] = cT[noff[q] + t];
            #pragma unroll
            for (int q = 0; q < 8; ++q) h[q] = __builtin_fmaf(e[q], h[q], u[q]);
            // Readout into column j of the 16d x 16t output tile:
            // 4 chained K=4 WMMAs; B operand is C_t masked to column j.
            const bool sel = (row == j);
            #pragma unroll
            for (int g = 0; g < 4; ++g) {
                v2f av; av.x = h[2 * g];              av.y = h[2 * g + 1];
                v2f bv; bv.x = sel ? cv[2 * g] : 0.f; bv.y = sel ? cv[2 * g + 1] : 0.f;
                if (j & 1)
                    acc1 = __builtin_amdgcn_wmma_f32_16x16x4_f32(
                        false, av, false, bv, (short)0, acc1, false, false);
                else
                    acc0 = __builtin_amdgcn_wmma_f32_16x16x4_f32(
                        false, av, false, bv, (short)0, acc0, false, false);
            }
        }
        // D-layout: lane col = row (=t index), VGPR r -> d = d0 + r + 8*hi
        const int tcol = cbase + sub * 16 + row;
        #pragma unroll
        for (int r = 0; r < 8; ++r) {
            size_t idx = obase + (size_t)(d0 + r + 8 * hi) * LSEQ + tcol;
            float xv2 = xT[(r + 8 * hi) * ROWP + sub * 16 + row];
            float yv  = (acc0[r] + acc1[r]) + xv2 * Dv[r];
            float zv  = z[idx];
            float sig = 1.f / (1.f + __builtin_amdgcn_exp2f(-zv * 1.44269504088896340736f));
            out[idx] = yv * (zv * sig);
        }
    }
}

// =============================================================================
extern "C" void kernel_launch(void* const* d_in, const int* in_sizes, int n_in,
                              void* d_out, int out_size, void* d_ws, size_t ws_size,
                              hipStream_t stream) {
    const float* x     = (const float*)d_in[0];
    const float* delta = (const float*)d_in[1];
    const float* A     = (const float*)d_in[2];
    const float* Bm    = (const float*)d_in[3];
    const float* Cm    = (const float*)d_in[4];
    const float* Dp    = (const float*)d_in[5];
    const float* z     = (const float*)d_in[6];
    float* out = (float*)d_out;

    float* aBuf = (float*)d_ws;                        // aProd, then hIn (in place)
    const size_t S = (size_t)BSZ * DBLKS * NC * 256;   // 1.57M floats
    float* hBuf = aBuf + S;

    dim3 blk(128, 1, 1);
    dim3 grd(NC, DBLKS / WPB, BSZ);                    // 32 x 24 x 2
    ssm_phaseA<<<grd, blk, 0, stream>>>(x, delta, A, Bm, aBuf, hBuf);
    ssm_phaseB<<<dim3((BSZ * DBLKS * 256) / 256, 1, 1), dim3(256, 1, 1), 0, stream>>>(aBuf, hBuf);
    ssm_phaseC<<<grd, blk, 0, stream>>>(x, delta, A, Bm, Cm, Dp, z, aBuf, out);
}